// SelfAttention_21852793602051
// MI455X (gfx1250) — compile-verified
//
#include <hip/hip_runtime.h>
#include <math.h>

// CDNA5 (gfx1250) wave32 fused self-attention, fp32 throughout using
// V_WMMA_F32_16X16X4_F32 (exact fp32 matrix path; reference is fp32).
//
// EMB=1024, K=64, H=8, B=4, T=2048.

typedef float v2f __attribute__((ext_vector_type(2)));
typedef float v8f __attribute__((ext_vector_type(8)));

#define EMB 1024
#define HEADS 8
#define HK 64
#define BATCH 4
#define SEQ 2048
#define NH (HEADS * HK)   // 512
#define ROWS (BATCH * SEQ) // 8192

__device__ __forceinline__ v8f wmma_f32(v2f a, v2f b, v8f c) {
  // (neg_a, A, neg_b, B, c_mod, C, reuse_a, reuse_b)
  return __builtin_amdgcn_wmma_f32_16x16x4_f32(false, a, false, b, (short)0, c,
                                               false, false);
}

// ---------------------------------------------------------------------------
// Kernel 1: QKV projection.  out[(b*H+h), t, kk] = (x[b,t,:] @ W[:, h*64+kk]) * scale
// One wave computes a 16x64 output tile; K-loop in steps of 4.
// A frag (16x4 f32): lanes 0-15 rows M=0..15 (k0,k0+1); lanes 16-31 same rows (k0+2,k0+3).
// B frag (4x16 f32): lanes 0-15 col N=lane (k0,k0+1); lanes 16-31 col N=lane-16 (k0+2,k0+3).
// C/D: vgpr j -> row j (lanes 0-15, col=lane) / row j+8 (lanes 16-31, col=lane-16).
// ---------------------------------------------------------------------------
__global__ void qkv_kernel(const float* __restrict__ x,
                           const float* __restrict__ Wq,
                           const float* __restrict__ Wk,
                           const float* __restrict__ Wv,
                           float* __restrict__ qo, float* __restrict__ ko,
                           float* __restrict__ vo) {
  const int lane = threadIdx.x;
  const int l15 = lane & 15;
  const int half = lane >> 4;

  const float qk_scale = 0.35355339059327373f; // 1 / 64^0.25
  const float* W;
  float* out;
  float scale;
  if (blockIdx.z == 0)      { W = Wq; out = qo; scale = qk_scale; }
  else if (blockIdx.z == 1) { W = Wk; out = ko; scale = qk_scale; }
  else                      { W = Wv; out = vo; scale = 1.0f; }

  const int row0 = blockIdx.x * 16;      // 0..8176
  const int n0 = blockIdx.y * 64;        // 0..448

  v8f acc0 = {}, acc1 = {}, acc2 = {}, acc3 = {};
  const float* xrow = x + (size_t)(row0 + l15) * EMB;

  for (int k0 = 0; k0 < EMB; k0 += 4) {
    v2f a = *(const v2f*)&xrow[k0 + half * 2];
    const float* w0 = W + (size_t)(k0 + half * 2) * NH + n0 + l15;
#pragma unroll
    for (int nt = 0; nt < 4; ++nt) {
      v2f bfrag;
      bfrag.x = w0[nt * 16];
      bfrag.y = w0[nt * 16 + NH];
      if (nt == 0) acc0 = wmma_f32(a, bfrag, acc0);
      else if (nt == 1) acc1 = wmma_f32(a, bfrag, acc1);
      else if (nt == 2) acc2 = wmma_f32(a, bfrag, acc2);
      else acc3 = wmma_f32(a, bfrag, acc3);
    }
  }

  v8f accs[4] = {acc0, acc1, acc2, acc3};
#pragma unroll
  for (int nt = 0; nt < 4; ++nt) {
#pragma unroll
    for (int jj = 0; jj < 8; ++jj) {
      int rl = jj + half * 8;
      int grow = row0 + rl;
      int b = grow >> 11;        // /SEQ
      int t = grow & (SEQ - 1);
      int n = n0 + nt * 16 + l15;
      int h = n >> 6;
      int kk = n & 63;
      out[(((size_t)(b * HEADS + h)) * SEQ + t) * HK + kk] = accs[nt][jj] * scale;
    }
  }
}

// ---------------------------------------------------------------------------
// Kernel 2: flash attention, one wave per (head, 16-query tile).
// Online softmax; P re-fragmented via padded LDS tile; masking only for
// flattened heads 0..3 (reference masks dot[:b] only).
// ---------------------------------------------------------------------------
__global__ void attn_kernel(const float* __restrict__ q,
                            const float* __restrict__ kws,
                            const float* __restrict__ vws,
                            const int* __restrict__ pad,
                            float* __restrict__ aout) {
  const int lane = threadIdx.x;
  const int l15 = lane & 15;
  const int half = lane >> 4;
  const int bh = blockIdx.x;            // 0..31
  const int i0 = blockIdx.y * 16;       // query tile base

  __shared__ float lds_p[16 * 18];      // padded stride 18: conflict-free, 8B aligned

  // Preload Q fragments for all 16 k-steps (K=64 -> 16 steps of 4)
  v2f qa[16];
  const float* qrow = q + ((size_t)bh * SEQ + i0 + l15) * HK;
#pragma unroll
  for (int s = 0; s < 16; ++s) qa[s] = *(const v2f*)&qrow[4 * s + half * 2];

  const bool masked = (bh < BATCH);
  const int thresh = masked ? (SEQ - pad[bh]) : 0x7fffffff;

  float m[8], l[8];
  v8f acc0 = {}, acc1 = {}, acc2 = {}, acc3 = {};
#pragma unroll
  for (int jj = 0; jj < 8; ++jj) { m[jj] = -INFINITY; l[jj] = 0.0f; }

  const float* kbase = kws + (size_t)bh * SEQ * HK;
  const float* vbase = vws + (size_t)bh * SEQ * HK;

  for (int j0 = 0; j0 < SEQ; j0 += 16) {
    // Prefetch next tile's K/V rows into cache while this tile computes
    // (emits global_prefetch_b8; predictable streaming access).
    if (j0 + 16 < SEQ) {
      __builtin_prefetch(&kbase[(size_t)(j0 + 16 + l15) * HK], 0, 3);
      __builtin_prefetch(&vbase[(size_t)(j0 + 16 + l15) * HK], 0, 3);
    }

    // ---- S = Q * K^T for this 16x16 tile
    v8f sv = {};
    const float* krow = kbase + (size_t)(j0 + l15) * HK;
#pragma unroll
    for (int s = 0; s < 16; ++s) {
      v2f kb = *(const v2f*)&krow[4 * s + half * 2];
      sv = wmma_f32(qa[s], kb, sv);
    }

    if (masked) {
      int gj = j0 + l15;
#pragma unroll
      for (int jj = 0; jj < 8; ++jj) {
        int gi = i0 + jj + half * 8;
        if (gi >= thresh && gj >= thresh) sv[jj] = -INFINITY;
      }
    }

    // ---- online softmax update (row r = jj + half*8 lives in vgpr jj, 16-lane half)
#pragma unroll
    for (int jj = 0; jj < 8; ++jj) {
      float rm = sv[jj];
      rm = fmaxf(rm, __shfl_xor(rm, 1, 32));
      rm = fmaxf(rm, __shfl_xor(rm, 2, 32));
      rm = fmaxf(rm, __shfl_xor(rm, 4, 32));
      rm = fmaxf(rm, __shfl_xor(rm, 8, 32));
      float mn = fmaxf(m[jj], rm);
      float sc = __expf(m[jj] - mn);   // 0 on first finite tile (m = -inf)
      float pv = __expf(sv[jj] - mn);  // 0 where masked
      float rs = pv;
      rs += __shfl_xor(rs, 1, 32);
      rs += __shfl_xor(rs, 2, 32);
      rs += __shfl_xor(rs, 4, 32);
      rs += __shfl_xor(rs, 8, 32);
      l[jj] = l[jj] * sc + rs;
      m[jj] = mn;
      acc0[jj] *= sc; acc1[jj] *= sc; acc2[jj] *= sc; acc3[jj] *= sc;
      lds_p[(jj + half * 8) * 18 + l15] = pv;  // P tile, row-major padded
    }
    __syncthreads();

    // ---- acc += P (16x16) * V (16x64)
#pragma unroll
    for (int s = 0; s < 4; ++s) {
      v2f pa = *(const v2f*)&lds_p[l15 * 18 + 4 * s + half * 2]; // A frag of P
      const float* vrow = vbase + (size_t)(j0 + 4 * s + half * 2) * HK + l15;
#pragma unroll
      for (int nt = 0; nt < 4; ++nt) {
        v2f vb;
        vb.x = vrow[nt * 16];
        vb.y = vrow[nt * 16 + HK];
        if (nt == 0) acc0 = wmma_f32(pa, vb, acc0);
        else if (nt == 1) acc1 = wmma_f32(pa, vb, acc1);
        else if (nt == 2) acc2 = wmma_f32(pa, vb, acc2);
        else acc3 = wmma_f32(pa, vb, acc3);
      }
    }
    __syncthreads();
  }

  // ---- finalize: divide by l, store in (b, t, h*64+kk) layout
  const int b = bh >> 3;
  const int h = bh & 7;
  v8f accs[4] = {acc0, acc1, acc2, acc3};
#pragma unroll
  for (int jj = 0; jj < 8; ++jj) {
    float inv = 1.0f / l[jj];
    int t = i0 + jj + half * 8;
#pragma unroll
    for (int nt = 0; nt < 4; ++nt) {
      int col = h * HK + nt * 16 + l15;
      aout[((size_t)(b * SEQ + t)) * NH + col] = accs[nt][jj] * inv;
    }
  }
}

// ---------------------------------------------------------------------------
// Kernel 3: output projection  (8192 x 512) @ (512 x 64) + bias
// ---------------------------------------------------------------------------
__global__ void proj_kernel(const float* __restrict__ a,
                            const float* __restrict__ Wu,
                            const float* __restrict__ bu,
                            float* __restrict__ out) {
  const int lane = threadIdx.x;
  const int l15 = lane & 15;
  const int half = lane >> 4;
  const int row0 = blockIdx.x * 16;

  v8f acc0 = {}, acc1 = {}, acc2 = {}, acc3 = {};
  const float* arow = a + (size_t)(row0 + l15) * NH;

  for (int k0 = 0; k0 < NH; k0 += 4) {
    v2f af = *(const v2f*)&arow[k0 + half * 2];
    const float* w0 = Wu + (size_t)(k0 + half * 2) * HK + l15;
#pragma unroll
    for (int nt = 0; nt < 4; ++nt) {
      v2f bfrag;
      bfrag.x = w0[nt * 16];
      bfrag.y = w0[nt * 16 + HK];
      if (nt == 0) acc0 = wmma_f32(af, bfrag, acc0);
      else if (nt == 1) acc1 = wmma_f32(af, bfrag, acc1);
      else if (nt == 2) acc2 = wmma_f32(af, bfrag, acc2);
      else acc3 = wmma_f32(af, bfrag, acc3);
    }
  }

  v8f accs[4] = {acc0, acc1, acc2, acc3};
#pragma unroll
  for (int nt = 0; nt < 4; ++nt) {
    float bias = bu[nt * 16 + l15];
#pragma unroll
    for (int jj = 0; jj < 8; ++jj) {
      int r = row0 + jj + half * 8;
      out[(size_t)r * HK + nt * 16 + l15] = accs[nt][jj] + bias;
    }
  }
}

// ---------------------------------------------------------------------------
extern "C" void kernel_launch(void* const* d_in, const int* in_sizes, int n_in,
                              void* d_out, int out_size, void* d_ws, size_t ws_size,
                              hipStream_t stream) {
  const float* x  = (const float*)d_in[0];
  const int* pad  = (const int*)d_in[1];
  const float* Wq = (const float*)d_in[2];
  const float* Wk = (const float*)d_in[3];
  const float* Wv = (const float*)d_in[4];
  const float* Wu = (const float*)d_in[5];
  const float* bu = (const float*)d_in[6];
  float* out = (float*)d_out;

  float* ws = (float*)d_ws;
  const size_t per = (size_t)BATCH * HEADS * SEQ * HK;  // 4,194,304 floats
  float* qbuf = ws;
  float* kbuf = ws + per;
  float* vbuf = ws + 2 * per;
  float* abuf = ws + 3 * per;  // (B*T, 512)

  // QKV projections: 512 row-tiles x 8 col64-tiles x {q,k,v}
  qkv_kernel<<<dim3(ROWS / 16, NH / 64, 3), dim3(32), 0, stream>>>(
      x, Wq, Wk, Wv, qbuf, kbuf, vbuf);

  // Flash attention: 32 heads x 128 query tiles
  attn_kernel<<<dim3(BATCH * HEADS, SEQ / 16), dim3(32), 0, stream>>>(
      qbuf, kbuf, vbuf, pad, abuf);

  // Output projection
  proj_kernel<<<dim3(ROWS / 16), dim3(32), 0, stream>>>(abuf, Wu, bu, out);
}